// LogisticNet_33200097198823
// MI455X (gfx1250) — compile-verified
//
#include <hip/hip_runtime.h>

// MI455X / gfx1250, wave32. Fused maxpool3d(2) + int4-dequant GEMM(2048->4) + softmax.
// Bandwidth bound: 512MB x read once ~= 22us floor @ 23.3 TB/s. GEMM runs on
// v_wmma_f32_16x16x4_f32 (exact f32), K split across 4 waves per 16-batch tile.

typedef __attribute__((ext_vector_type(2))) float v2f;
typedef __attribute__((ext_vector_type(8))) float v8f;

#define THREADS 256
#define WAVES 8
#define BATCH_PER_BLOCK 32
#define FEATS 2048

__device__ __forceinline__ float4 f4max(float4 a, float4 b) {
    return make_float4(fmaxf(a.x, b.x), fmaxf(a.y, b.y),
                       fmaxf(a.z, b.z), fmaxf(a.w, b.w));
}

__global__ __launch_bounds__(THREADS)
void logistic_fused_kernel(const float* __restrict__ x,
                           const float* __restrict__ W,
                           const float* __restrict__ bias,
                           float* __restrict__ out)
{
    __shared__ float s_wq[FEATS * 4];       // 32KB: WqT[f][o] dequantized int4 weight
    __shared__ float s_pool[WAVES * 512];   // 16KB: per-wave pooled chunk [16 m][32 f]
    __shared__ float s_part[WAVES * 256];   // 8KB : per-wave 16x16 partial C tile
    __shared__ float s_red[THREADS];        // 1KB : absmax reduction

    const int t = threadIdx.x;

    // ---- Phase 1: per-tensor symmetric int4 quantize W, store transposed in LDS ----
    float mx = 0.0f;
    for (int i = t; i < FEATS * 4; i += THREADS) mx = fmaxf(mx, fabsf(W[i]));
    s_red[t] = mx;
    __syncthreads();
    for (int s2 = THREADS / 2; s2 > 0; s2 >>= 1) {
        if (t < s2) s_red[t] = fmaxf(s_red[t], s_red[t + s2]);
        __syncthreads();
    }
    const float scale = s_red[0] * (1.0f / 7.0f);
    const float inv = (scale != 0.0f) ? 1.0f / scale : 0.0f;
    for (int i = t; i < FEATS * 4; i += THREADS) {
        int o = i >> 11;            // output class 0..3 (row-major W[4][2048])
        int f = i & (FEATS - 1);
        float q = rintf(W[i] * inv);            // round-half-even == jnp.round
        q = fminf(fmaxf(q, -8.0f), 7.0f);       // clip to int4 range
        s_wq[f * 4 + o] = q * scale;
    }
    __syncthreads();

    // ---- Phase 2: fused 2x2x2 maxpool + WMMA partial GEMM ----
    const int wave = t >> 5;
    const int lane = t & 31;
    const int tile = wave >> 2;     // 0/1: which 16-batch tile of this block
    const int cc   = wave & 3;      // channel == K-quarter (512 features) for this wave
    const int m    = lane >> 1;     // batch-in-tile handled during pooling
    const int hsel = lane & 1;      // which h'-pair this lane pools
    const int bBase = blockIdx.x * BATCH_PER_BLOCK + tile * 16;

    const float* xb = x + (size_t)(bBase + m) * 16384 + (size_t)cc * 4096;
    float* s_mypool = &s_pool[wave * 512];

    v8f acc = {0.f, 0.f, 0.f, 0.f, 0.f, 0.f, 0.f, 0.f};
    const int m_a  = lane & 15;         // A-fragment row (batch)
    const int koff = (lane >> 4) * 2;   // A layout: lanes 16-31 hold K+2,K+3
    const int n    = lane & 15;         // B/C column

    for (int dp = 0; dp < 8; ++dp) {            // pooled depth index
        const float* xd = xb + dp * 512;        // two input d-slices
        for (int hh = 0; hh < 2; ++hh) {        // half of the h'-plane (32 features)
            // pooling: each lane produces 2 rows of 8 pooled values
            #pragma unroll
            for (int hp = 0; hp < 2; ++hp) {
                int hl  = hsel * 2 + hp;        // local h' 0..3
                int hpr = hh * 4 + hl;          // global h' 0..7
                const float4* r00 = (const float4*)(xd +       (2 * hpr) * 16);
                const float4* r01 = (const float4*)(xd +       (2 * hpr + 1) * 16);
                const float4* r10 = (const float4*)(xd + 256 + (2 * hpr) * 16);
                const float4* r11 = (const float4*)(xd + 256 + (2 * hpr + 1) * 16);
                float p[8];
                #pragma unroll
                for (int w4 = 0; w4 < 4; ++w4) {
                    float4 v = f4max(f4max(r00[w4], r01[w4]),
                                     f4max(r10[w4], r11[w4]));
                    p[2 * w4]     = fmaxf(v.x, v.y);
                    p[2 * w4 + 1] = fmaxf(v.z, v.w);
                }
                float4* dst = (float4*)&s_mypool[m * 32 + hl * 8];
                dst[0] = make_float4(p[0], p[1], p[2], p[3]);
                dst[1] = make_float4(p[4], p[5], p[6], p[7]);
            }
            // 8 WMMA K-steps (K=4 each) consume the 32 pooled features
            const int fBase = cc * 512 + dp * 64 + hh * 32;
            #pragma unroll
            for (int kk = 0; kk < 8; ++kk) {
                // A: 16x4 f32 fragment — lane m_a holds K={koff,koff+1}
                v2f a = *(const v2f*)&s_mypool[m_a * 32 + kk * 4 + koff];
                // B: 4x16 f32 fragment — VGPR0 rows K=0/1, VGPR1 rows K=2/3
                v2f bfrag;
                #pragma unroll
                for (int j = 0; j < 2; ++j) {
                    int k = 2 * j + (lane >> 4);
                    bfrag[j] = (n < 4) ? s_wq[(fBase + kk * 4 + k) * 4 + n] : 0.0f;
                }
                acc = __builtin_amdgcn_wmma_f32_16x16x4_f32(
                        false, a, false, bfrag, (short)0, acc, false, false);
            }
        }
    }

    // write per-wave partial C tile (16 batches x 16 cols, cols 4..15 are zero)
    #pragma unroll
    for (int j = 0; j < 8; ++j) {
        int M = j + (lane >> 4) * 8;
        s_part[wave * 256 + M * 16 + n] = acc[j];
    }
    __syncthreads();

    // ---- Phase 3: reduce K-partials across 4 waves, bias, softmax, store ----
    if (t < BATCH_PER_BLOCK) {
        int tl = t >> 4, mm = t & 15;
        float l[4];
        #pragma unroll
        for (int o = 0; o < 4; ++o) {
            float s = 0.f;
            #pragma unroll
            for (int w = 0; w < 4; ++w)
                s += s_part[(tl * 4 + w) * 256 + mm * 16 + o];
            l[o] = s + bias[o];
        }
        float lm = fmaxf(fmaxf(l[0], l[1]), fmaxf(l[2], l[3]));
        float e0 = __expf(l[0] - lm), e1 = __expf(l[1] - lm);
        float e2 = __expf(l[2] - lm), e3 = __expf(l[3] - lm);
        float r = 1.0f / (e0 + e1 + e2 + e3);
        float4* po = (float4*)(out + (size_t)(blockIdx.x * BATCH_PER_BLOCK + t) * 4);
        *po = make_float4(e0 * r, e1 * r, e2 * r, e3 * r);
    }
}

extern "C" void kernel_launch(void* const* d_in, const int* in_sizes, int n_in,
                              void* d_out, int out_size, void* d_ws, size_t ws_size,
                              hipStream_t stream) {
    (void)n_in; (void)out_size; (void)d_ws; (void)ws_size;
    const float* x    = (const float*)d_in[0];   // [B,4,16,16,16] f32
    const float* W    = (const float*)d_in[1];   // [4,2048] f32
    const float* bias = (const float*)d_in[2];   // [4] f32
    float* out = (float*)d_out;                  // [B,4] f32

    int B = in_sizes[0] / 16384;                 // 8192
    int blocks = B / BATCH_PER_BLOCK;            // 256 blocks x 256 threads
    logistic_fused_kernel<<<blocks, THREADS, 0, stream>>>(x, W, bias, out);
}